// CentroidDistance_57655640981898
// MI455X (gfx1250) — compile-verified
//
#include <hip/hip_runtime.h>
#include <hip/hip_bf16.h>

typedef __attribute__((ext_vector_type(2))) float v2f;
typedef __attribute__((ext_vector_type(8))) float v8f;

#define TILE  16
#define DIM   128      // feature dim (fixed by reference)
#define KCENT 128      // number of centroids (fixed by reference)
#define LDA   132      // padded LDS row stride (132 % 64 == 4 -> conflict-free)

// Each block: 16 node rows x all 128 centroid columns.
// 8 waves; wave w computes the 16x16 tile for centroid columns [16w, 16w+16).
// GEMM via V_WMMA_F32_16X16X4_F32, 32 k-steps over d=128.
__global__ __launch_bounds__(256)
void CentroidDistance_gemm_kernel(const float* __restrict__ node,
                                  const float* __restrict__ mask,
                                  const float* __restrict__ cent,
                                  float* __restrict__ out_node,   // [N, KCENT]
                                  float* __restrict__ partial,    // [gridDim.x, KCENT]
                                  int N)
{
    __shared__ float sA[TILE * LDA];
    __shared__ float s_x2[TILE];
    __shared__ float s_y2[KCENT];

    const int tid  = threadIdx.x;
    const int wave = tid >> 5;
    const int lane = tid & 31;
    const int nb   = blockIdx.x * TILE;   // first node row of this block
    const int cb   = wave * TILE;         // first centroid column of this wave

    // ---- stage node tile (16 x 128 f32) into LDS, float4 loads ----
    for (int i = tid; i < TILE * DIM / 4; i += 256) {
        const int r = i >> 5;             // 32 float4 per row
        const int c = (i & 31) * 4;
        const float4 v = *(const float4*)(node + (size_t)(nb + r) * DIM + c);
        sA[r * LDA + c + 0] = v.x;
        sA[r * LDA + c + 1] = v.y;
        sA[r * LDA + c + 2] = v.z;
        sA[r * LDA + c + 3] = v.w;
    }
    __syncthreads();

    // ---- squared norms: y2 for all centroids (global, L2-hot), x2 from LDS ----
    if (tid < KCENT) {
        const float* cp = cent + (size_t)tid * DIM;
        float acc = 0.f;
        #pragma unroll 4
        for (int j = 0; j < DIM; ++j) acc = fmaf(cp[j], cp[j], acc);
        s_y2[tid] = acc;
    } else if (tid < KCENT + TILE) {
        const int r = tid - KCENT;
        float acc = 0.f;
        #pragma unroll 4
        for (int j = 0; j < DIM; ++j) { const float v = sA[r * LDA + j]; acc = fmaf(v, v, acc); }
        s_x2[r] = acc;
    }
    __syncthreads();

    // ---- WMMA f32 GEMM: xy[16x16] = nodeTile (16xD) * centTile^T (Dx16) ----
    const int m    = lane & 15;           // A: row M / B: col N
    const int koff = (lane >> 4) * 2;     // K sub-offset 0 or 2 per half-wave
    const float* bptr = cent + (size_t)(cb + m) * DIM + koff;
    const float* aptr = sA + m * LDA + koff;

    v8f acc = {0.f, 0.f, 0.f, 0.f, 0.f, 0.f, 0.f, 0.f};
    #pragma unroll
    for (int kk = 0; kk < DIM; kk += 4) {
        const v2f a = *(const v2f*)(aptr + kk);   // ds_load_b64, conflict-free
        const v2f b = *(const v2f*)(bptr + kk);   // global_load_b64, L2-resident
        acc = __builtin_amdgcn_wmma_f32_16x16x4_f32(
                  false, a, false, b, (short)0, acc, false, false);
    }

    // ---- epilogue: poincare sqdist from scalars (c = 1) ----
    // C/D layout: lanes 0-15 -> N=lane, M=r; lanes 16-31 -> N=lane-16, M=8+r
    const int mbase = (lane >> 4) << 3;   // 0 or 8
    const int kcol  = cb + m;
    const float y2  = s_y2[kcol];

    float colsum = 0.f;
    #pragma unroll
    for (int r = 0; r < 8; ++r) {
        const int   nidx = nb + mbase + r;
        const float s    = acc[r];            // <x, y>
        const float x2   = s_x2[mbase + r];
        const float uv   = -s;                // <-x, y>
        const float p    = fmaf(2.f, uv, 1.f);        // 1 + 2c*uv
        const float Aq   = p + y2;                    // 1 + 2c*uv + c*y2
        const float Bq   = 1.f - x2;                  // 1 - c*x2
        const float den  = fmaf(x2, y2, p);           // 1 + 2c*uv + c^2*x2*y2
        float num2 = Aq * Aq * x2 + 2.f * Aq * Bq * uv + Bq * Bq * y2;
        num2 = fmaxf(num2, 0.f);
        const float denc = fmaxf(den, 1e-15f);
        float t = __builtin_sqrtf(num2) / denc;       // sqrt(c) * ||diff||
        t = fminf(t, 1.f - 1e-7f);                    // artanh clip (t >= 0)
        const float dist = log1pf(2.f * t / (1.f - t));   // 2/sqrt(c)*artanh(t)
        float ncd = __builtin_sqrtf(fmaxf(dist * dist, 1e-15f));
        ncd *= mask[nidx];
        out_node[(size_t)nidx * KCENT + kcol] = ncd;
        colsum += ncd;
    }

    // combine the two half-wave row groups (M 0-7 with M 8-15), same kcol
    colsum += __shfl_xor(colsum, 16, 32);
    if (lane < 16)
        partial[(size_t)blockIdx.x * KCENT + kcol] = colsum;
}

// Deterministic final reduction: graph[k] = sum_b partial[b][k] / sum(mask)
__global__ __launch_bounds__(128)
void CentroidDistance_reduce_kernel(const float* __restrict__ partial,
                                    const float* __restrict__ mask,
                                    float* __restrict__ out_graph,  // [KCENT]
                                    int numBlocks, int N)
{
    __shared__ float sm[128];
    const int t = threadIdx.x;

    float macc = 0.f;
    for (int i = t; i < N; i += 128) macc += mask[i];
    sm[t] = macc;
    __syncthreads();
    #pragma unroll
    for (int s = 64; s > 0; s >>= 1) {
        if (t < s) sm[t] += sm[t + s];
        __syncthreads();
    }
    const float msum = sm[0];

    float acc = 0.f;
    for (int b = 0; b < numBlocks; ++b)
        acc += partial[(size_t)b * KCENT + t];
    out_graph[t] = acc / msum;
}

extern "C" void kernel_launch(void* const* d_in, const int* in_sizes, int n_in,
                              void* d_out, int out_size, void* d_ws, size_t ws_size,
                              hipStream_t stream)
{
    const float* node = (const float*)d_in[0];   // [N, 128]
    const float* mask = (const float*)d_in[1];   // [N, 1]
    const float* cent = (const float*)d_in[2];   // [128, 128]
    float* out = (float*)d_out;                  // [128] graph ++ [N*128] node dists

    const int N = in_sizes[0] / DIM;             // 8192
    const int numBlocks = N / TILE;              // 512
    float* partial = (float*)d_ws;               // numBlocks * 128 floats (256 KB)

    CentroidDistance_gemm_kernel<<<numBlocks, 256, 0, stream>>>(
        node, mask, cent, out + KCENT, partial, N);
    CentroidDistance_reduce_kernel<<<1, 128, 0, stream>>>(
        partial, mask, out, numBlocks, N);
}